// GL_48515950576089
// MI455X (gfx1250) — compile-verified
//
#include <hip/hip_runtime.h>
#include <hip/hip_bf16.h>
#include <math.h>

typedef float v2f __attribute__((ext_vector_type(2)));
typedef float v8f __attribute__((ext_vector_type(8)));

// ---------------------------------------------------------------------------
// init: x1 = 0, x21 = b2, x2 = b3, t = xTB - x
// ---------------------------------------------------------------------------
__global__ void init_kernel(const float* __restrict__ b2,
                            const float* __restrict__ b3,
                            const float* __restrict__ x,
                            const float* __restrict__ xTB,
                            float* __restrict__ x1,
                            float* __restrict__ x21,
                            float* __restrict__ x2,
                            float* __restrict__ t_vec, int n) {
    int i = blockIdx.x * blockDim.x + threadIdx.x;
    if (i < n) {
        x1[i]    = 0.0f;
        x21[i]   = b2[i];
        x2[i]    = b3[i];
        t_vec[i] = xTB[i] - x[i];
    }
}

// ---------------------------------------------------------------------------
// GNN1 edge scatter: x1[dst] += (x[src] - x[dst]) * att[e]
// ---------------------------------------------------------------------------
__global__ void edge_kernel(const float* __restrict__ x,
                            const int* __restrict__ edge_index,  // [2,E] row-major
                            const float* __restrict__ att,
                            float* __restrict__ x1, int E) {
    int e = blockIdx.x * blockDim.x + threadIdx.x;
    if (e < E) {
        int s = edge_index[e];      // row 0: src
        int d = edge_index[E + e];  // row 1: dst
        float m = (x[s] - x[d]) * att[e];
        atomicAdd(&x1[d], m);
    }
}

// ---------------------------------------------------------------------------
// GEMV via V_WMMA_F32_16X16X4_F32.
//   out[rowBase..rowBase+15] += W[16 rows, kSlab] @ v[kSlab]
// Grid: (KSLABS, N/16), 32 threads (one wave32) per block.
// A layout (16x4 f32): lanes 0-15 hold row M=lane,    VGPR0/1 = K{0,1};
//                      lanes 16-31 hold row M=lane-16, VGPR0/1 = K{2,3}.
// B is filled with v replicated into all 16 columns, so every D column equals
// the GEMV result; lanes 0 and 16 commit rows 0-7 / 8-15 of the tile.
// ---------------------------------------------------------------------------
__global__ void gemv_wmma_kernel(const float* __restrict__ W,
                                 const float* __restrict__ vec,
                                 float* __restrict__ out,
                                 int n) {
    const int lane    = threadIdx.x;            // wave32
    const int rowBase = blockIdx.y * 16;
    const int kLen    = n / gridDim.x;
    const int kBase   = blockIdx.x * kLen;
    const int kEnd    = kBase + kLen;

    const int row = rowBase + (lane & 15);
    const float* __restrict__ wrow = W + (long)row * n;
    const int halfOff = (lane >> 4) << 1;       // 0 for lanes 0-15, 2 for 16-31

    v8f c = {};

#pragma unroll 4
    for (int k = kBase; k < kEnd; k += 4) {
        const int kc = k + halfOff;
        // A fragment: two consecutive W elements of this lane's row (b64)
        v2f a = *(const v2f*)(wrow + kc);
        // B fragment: matching vector elements, replicated across columns (b64)
        v2f b = *(const v2f*)(vec + kc);
        // D = A*B + C, f32 16x16x4 (neg_a, A, neg_b, B, c_mod, C, reuse_a, reuse_b)
        c = __builtin_amdgcn_wmma_f32_16x16x4_f32(false, a, false, b,
                                                  (short)0, c, false, false);
    }

    // Commit: lane 0 holds rows rowBase+0..7 in c[0..7]; lane 16 rows +8..15.
    if (lane == 0 || lane == 16) {
        int r0 = rowBase + ((lane >> 4) << 3);
#pragma unroll
        for (int r = 0; r < 8; ++r) {
            atomicAdd(&out[r0 + r], c[r]);
        }
    }
}

// ---------------------------------------------------------------------------
// LSTM scan (T steps, H=8, input_size=1) fused with W1 dot product.
// One wave32: lane l owns gate row l of the 4H=32 gates (order i,f,g,o).
// Lanes replicate cell j = lane&7 state; h broadcast via shuffles.
// Writes scalar (W1 @ hs_flat + b1) to *out_scalar.
// ---------------------------------------------------------------------------
__global__ void lstm_kernel(const float* __restrict__ cur,
                            const float* __restrict__ W_ih,   // [32,1]
                            const float* __restrict__ W_hh,   // [32,8]
                            const float* __restrict__ b_ih,
                            const float* __restrict__ b_hh,
                            const float* __restrict__ W1,     // [1, T*8]
                            const float* __restrict__ b1,
                            float* __restrict__ out_scalar, int T) {
    const int lane = threadIdx.x;               // 0..31
    const int j    = lane & 7;                  // cell index this lane replicates

    const float wi   = W_ih[lane];
    const float bias = b_ih[lane] + b_hh[lane];
    float wr[8];
#pragma unroll
    for (int q = 0; q < 8; ++q) wr[q] = W_hh[lane * 8 + q];

    float h[8];
#pragma unroll
    for (int q = 0; q < 8; ++q) h[q] = 0.0f;
    float cst = 0.0f;                           // cell state for cell j
    float acc = 0.0f;                           // partial W1 dot (cell j)

    for (int t = 0; t < T; ++t) {
        float xt = cur[t];
        float gate = fmaf(wi, xt, bias);
#pragma unroll
        for (int q = 0; q < 8; ++q) gate = fmaf(wr[q], h[q], gate);

        // lanes 16-23 are the 'g' gate (tanh); others sigmoid
        float act = (lane >= 16 && lane < 24) ? tanhf(gate)
                                              : 1.0f / (1.0f + expf(-gate));

        float ig = __shfl(act, j,      32);
        float fg = __shfl(act, j + 8,  32);
        float gg = __shfl(act, j + 16, 32);
        float og = __shfl(act, j + 24, 32);

        cst = fmaf(fg, cst, ig * gg);
        float hj = og * tanhf(cst);

        // broadcast new h to every lane
#pragma unroll
        for (int q = 0; q < 8; ++q) h[q] = __shfl(hj, q, 32);

        acc = fmaf(W1[t * 8 + j], hj, acc);
    }

    // reduce acc over cells 0..7 (lanes 0..7 hold distinct cells)
    float total = 0.0f;
#pragma unroll
    for (int q = 0; q < 8; ++q) total += __shfl(acc, q, 32);

    if (lane == 0) *out_scalar = total + b1[0];
}

// ---------------------------------------------------------------------------
// final: out = x + x1 + x2 + scalar * x30
// ---------------------------------------------------------------------------
__global__ void final_kernel(const float* __restrict__ x,
                             const float* __restrict__ x1,
                             const float* __restrict__ x2,
                             const float* __restrict__ x30,
                             const float* __restrict__ scalar,
                             float* __restrict__ out, int n) {
    int i = blockIdx.x * blockDim.x + threadIdx.x;
    if (i < n) {
        out[i] = x[i] + x1[i] + x2[i] + (*scalar) * x30[i];
    }
}

// ---------------------------------------------------------------------------
extern "C" void kernel_launch(void* const* d_in, const int* in_sizes, int n_in,
                              void* d_out, int out_size, void* d_ws, size_t ws_size,
                              hipStream_t stream) {
    const float* x    = (const float*)d_in[0];
    const float* xTB  = (const float*)d_in[1];
    const float* cur  = (const float*)d_in[2];
    const int*   ei   = (const int*)  d_in[3];
    const float* att  = (const float*)d_in[4];
    const float* W2   = (const float*)d_in[5];
    const float* b2   = (const float*)d_in[6];
    const float* W3   = (const float*)d_in[7];
    const float* b3   = (const float*)d_in[8];
    const float* W_ih = (const float*)d_in[9];
    const float* W_hh = (const float*)d_in[10];
    const float* b_ih = (const float*)d_in[11];
    const float* b_hh = (const float*)d_in[12];
    const float* W1   = (const float*)d_in[13];
    const float* b1   = (const float*)d_in[14];
    const float* x30  = (const float*)d_in[15];

    const int N = in_sizes[0];          // 8192
    const int E = in_sizes[3] / 2;      // 262144
    const int T = in_sizes[2];          // 2048

    float* ws      = (float*)d_ws;
    float* x1_ws   = ws;                // [N]
    float* x21_ws  = ws + N;            // [N]
    float* x2_ws   = ws + 2 * N;        // [N]
    float* t_ws    = ws + 3 * N;        // [N]  (xTB - x)
    float* sc_ws   = ws + 4 * N;        // [1]

    float* out = (float*)d_out;

    const int TPB = 256;

    // 1) seed accumulators + precompute t = xTB - x
    init_kernel<<<(N + TPB - 1) / TPB, TPB, 0, stream>>>(b2, b3, x, xTB,
                                                         x1_ws, x21_ws, x2_ws, t_ws, N);

    // 2) GNN edge scatter
    edge_kernel<<<(E + TPB - 1) / TPB, TPB, 0, stream>>>(x, ei, att, x1_ws, E);

    // 3) LSTM head (independent data-wise; stream order is fine)
    lstm_kernel<<<1, 32, 0, stream>>>(cur, W_ih, W_hh, b_ih, b_hh, W1, b1, sc_ws, T);

    // 4) x21 = W2 @ t + b2   (WMMA GEMV, 8 K-slabs x 512 row-tiles)
    {
        dim3 grid(8, N / 16);
        gemv_wmma_kernel<<<grid, 32, 0, stream>>>(W2, t_ws, x21_ws, N);
    }

    // 5) x2 = W3 @ x21 + b3
    {
        dim3 grid(8, N / 16);
        gemv_wmma_kernel<<<grid, 32, 0, stream>>>(W3, x21_ws, x2_ws, N);
    }

    // 6) out = x + x1 + x2 + s * x30
    final_kernel<<<(N + TPB - 1) / TPB, TPB, 0, stream>>>(x, x1_ws, x2_ws, x30, sc_ws, out, N);
}